// MambaBlock_83468394430680
// MI455X (gfx1250) — compile-verified
//
#include <hip/hip_runtime.h>
#include <hip/hip_bf16.h>
#include <stdint.h>

// ---------------------------------------------------------------------------
// Mamba block forward for MI455X (gfx1250), fp32.
// Dense GEMMs: V_WMMA_F32_16X16X4_F32, LDS double-buffered pipeline, with
// direct global->LDS async copies (ASYNCcnt) when the toolchain exposes them.
// ---------------------------------------------------------------------------

typedef float v2f __attribute__((ext_vector_type(2)));
typedef float v8f __attribute__((ext_vector_type(8)));
typedef int   v4i_ __attribute__((vector_size(4 * sizeof(int))));

#define L_SEQ    2048
#define BATCH    2
#define D_MODEL  2048
#define D_INNER  4096
#define D_STATE  16
#define D_CONV   4
#define DT_RANK  128

#define KC       32            // K-chunk per LDS stage
#define LDS_STR  36            // padded row stride (floats): 144B, 16B-aligned, bank-spread

// ---- gfx1250 async global->LDS support (guarded; falls back to reg staging) ----
#if defined(__has_builtin)
#  if __has_builtin(__builtin_amdgcn_global_load_async_to_lds_b128)
#    define HAVE_ASYNC_LDS 1
#  else
#    define HAVE_ASYNC_LDS 0
#  endif
#  if HAVE_ASYNC_LDS && __has_builtin(__builtin_amdgcn_s_wait_asynccnt)
#    define HAVE_WAIT_ASYNC 1
#  else
#    define HAVE_WAIT_ASYNC 0
#  endif
#else
#  define HAVE_ASYNC_LDS 0
#  define HAVE_WAIT_ASYNC 0
#endif

typedef __attribute__((address_space(1))) v4i_* as1_v4i_ptr;   // global (__device__)
typedef __attribute__((address_space(3))) v4i_* as3_v4i_ptr;   // LDS   (__shared__)

__device__ __forceinline__ void async_copy16(const float* g, float* lds) {
#if HAVE_ASYNC_LDS
    __builtin_amdgcn_global_load_async_to_lds_b128(
        (as1_v4i_ptr)(uintptr_t)g,
        (as3_v4i_ptr)(uintptr_t)lds, 0, 0);
#else
    (void)g; (void)lds;
#endif
}

__device__ __forceinline__ void wait_async_all() {
#if HAVE_ASYNC_LDS
#  if HAVE_WAIT_ASYNC
    __builtin_amdgcn_s_wait_asynccnt(0);
#  else
    asm volatile("s_wait_asynccnt 0x0" ::: "memory");
#  endif
#endif
}

__device__ __forceinline__ v8f wmma_f32_k4(v2f a, v2f b, v8f c) {
    // D = A(16x4) * B(4x16) + C(16x16), all fp32.
    return __builtin_amdgcn_wmma_f32_16x16x4_f32(
        /*neg_a=*/false, a, /*neg_b=*/false, b,
        /*c_mod=*/(short)0, c, /*reuse_a=*/false, /*reuse_b=*/false);
}

// ---------------------------------------------------------------------------
// WMMA GEMM:  Out[b, m, n] (+epilogue) = sum_k A[m, k] * Bsrc_b[k, n]
//   A: row-major (lda), K-contiguous.
//   B element (k, n) at  Bsrc + b*bbs + k*ldbk + n*ldbn
//     BK=true : B contiguous along K (ldbk==1)  -> async/b128 loads along k
//     BK=false: B contiguous along N (ldbn==1)  -> b128 loads along n + LDS transpose
//   Out element (m, n) at Out + b*obs + m*oms + n*ons
// Block: 128 threads (4 waves), block tile 64(M) x 64(N), double-buffered LDS.
// Wave w owns M rows [w*16, w*16+16) x all 64 N columns (4 v8f accumulators).
// SP=true: v = softplus(v + bias[m])   (dt_proj path)
// ---------------------------------------------------------------------------
template <bool SP, bool BK>
__global__ void gemm_wmma_f32(const float* __restrict__ A,
                              const float* __restrict__ Bsrc,
                              float* __restrict__ Out,
                              const float* __restrict__ bias,
                              int M, int K,
                              long long lda,
                              long long bbs, long long ldbk, long long ldbn,
                              long long obs, long long oms, long long ons)
{
    const int tid  = threadIdx.x;
    const int lane = tid & 31;
    const int wave = tid >> 5;
    const int mblk = blockIdx.x * 64;
    const int nblk = blockIdx.y * 64;
    const int b    = blockIdx.z;

    const float* __restrict__ Bb = Bsrc + (long long)b * bbs;

    __shared__ __align__(16) float As[2][64 * LDS_STR];
    __shared__ __align__(16) float Bs[2][64 * LDS_STR];

    float4 ra[4], rb[4];

    // ---- stage K-chunk kt: async parts go straight to LDS buf, rest to regs ----
    auto issue_chunk = [&](int kt, int buf) {
        const int k0 = kt * KC;
#pragma unroll
        for (int j = 0; j < 4; ++j) {
            const int e = tid + 128 * j;        // 512 float4 per tile
            {   // A tile: 64 rows x 32 k  (8 float4 per row)
                const int row = e >> 3;
                const int kc  = (e & 7) * 4;
                int gr = mblk + row; if (gr > M - 1) gr = M - 1;   // ragged-M clamp
                const float* g = A + (long long)gr * lda + k0 + kc;
                if (HAVE_ASYNC_LDS) async_copy16(g, &As[buf][row * LDS_STR + kc]);
                else                ra[j] = *(const float4*)g;
            }
            if (BK) {   // B contiguous along k: rows n, 8 float4 of k each
                const int n  = e >> 3;
                const int kc = (e & 7) * 4;
                const float* g = Bb + (long long)(nblk + n) * ldbn + k0 + kc;
                if (HAVE_ASYNC_LDS) async_copy16(g, &Bs[buf][n * LDS_STR + kc]);
                else                rb[j] = *(const float4*)g;
            } else {    // B contiguous along n: 32 k-rows x 16 float4 of n
                const int kl = e >> 4;
                const int nc = (e & 15) * 4;
                rb[j] = *(const float4*)(Bb + (long long)(k0 + kl) * ldbk + nblk + nc);
            }
        }
    };

    // ---- commit the register-staged parts into LDS buffer `buf` ----
    auto commit_chunk = [&](int buf) {
#pragma unroll
        for (int j = 0; j < 4; ++j) {
            const int e = tid + 128 * j;
            if (!HAVE_ASYNC_LDS) {
                const int row = e >> 3;
                const int kc  = (e & 7) * 4;
                *(float4*)&As[buf][row * LDS_STR + kc] = ra[j];
                if (BK) *(float4*)&Bs[buf][row * LDS_STR + kc] = rb[j];
            }
            if (!BK) {  // transpose: scatter 4 consecutive n at fixed k
                const int kl = e >> 4;
                const int nc = (e & 15) * 4;
                Bs[buf][(nc + 0) * LDS_STR + kl] = rb[j].x;
                Bs[buf][(nc + 1) * LDS_STR + kl] = rb[j].y;
                Bs[buf][(nc + 2) * LDS_STR + kl] = rb[j].z;
                Bs[buf][(nc + 3) * LDS_STR + kl] = rb[j].w;
            }
        }
    };

    const int laneLo = lane & 15;
    const int kk     = (lane >> 4) * 2;       // K sub-offset per fp32 WMMA layout
    const int mrow   = wave * 16 + laneLo;    // block-local A row for this lane

    const v8f vzero = {0.f, 0.f, 0.f, 0.f, 0.f, 0.f, 0.f, 0.f};
    v8f acc[4] = {vzero, vzero, vzero, vzero};

    // ---- 32 WMMAs over one resident LDS chunk ----
    auto compute = [&](int buf) {
#pragma unroll
        for (int ks = 0; ks < KC; ks += 4) {
            const float* ap = &As[buf][mrow * LDS_STR + ks + kk];
            v2f a; a.x = ap[0]; a.y = ap[1];              // ds_load_b64
#pragma unroll
            for (int t = 0; t < 4; ++t) {
                const float* bp = &Bs[buf][(t * 16 + laneLo) * LDS_STR + ks + kk];
                v2f bf; bf.x = bp[0]; bf.y = bp[1];       // ds_load_b64
                acc[t] = wmma_f32_k4(a, bf, acc[t]);
            }
        }
    };

    // ---- double-buffered main loop (one barrier per chunk) ----
    const int nk = K / KC;
    issue_chunk(0, 0);
    commit_chunk(0);
    wait_async_all();
    __syncthreads();
    for (int kt = 0; kt < nk; ++kt) {
        const int buf = kt & 1;
        if (kt + 1 < nk) issue_chunk(kt + 1, buf ^ 1);  // overlaps compute
        compute(buf);
        if (kt + 1 < nk) {
            commit_chunk(buf ^ 1);   // safe: cross-wave reads of buf^1 ended pre-barrier
            wait_async_all();
            __syncthreads();
        }
    }

    // ---- epilogue + store. C/D layout: VGPR j holds rows (j, j+8); n = lane&15.
    float* __restrict__ Op = Out + (long long)b * obs;
    const int mhalf = (lane >> 4) * 8;
#pragma unroll
    for (int t = 0; t < 4; ++t) {
        const int n = nblk + t * 16 + laneLo;
#pragma unroll
        for (int j = 0; j < 8; ++j) {
            const int m = mblk + wave * 16 + mhalf + j;
            if (m < M) {
                float v = acc[t][j];
                if (SP) {
                    v += bias[m];
                    v = (v > 20.0f) ? v : log1pf(__expf(v));   // softplus
                }
                Op[(long long)m * oms + (long long)n * ons] = v;
            }
        }
    }
}

// ---------------------------------------------------------------------------
// Depthwise causal conv1d (K=4) + SiLU.  x layout: xz[b][d][l] (d < 4096 half).
// ---------------------------------------------------------------------------
__global__ void conv_silu_kernel(const float* __restrict__ xz,
                                 const float* __restrict__ conv_w,
                                 float* __restrict__ xc)
{
    const long long idx = (long long)blockIdx.x * blockDim.x + threadIdx.x;
    const long long total = (long long)BATCH * D_INNER * L_SEQ;
    if (idx >= total) return;

    const int l = (int)(idx % L_SEQ);
    const int d = (int)((idx / L_SEQ) % D_INNER);
    const int b = (int)(idx / ((long long)D_INNER * L_SEQ));

    const float* __restrict__ xrow =
        xz + ((long long)b * (2 * D_INNER) + d) * L_SEQ;
    const float* __restrict__ w = conv_w + d * D_CONV;

    float s = 0.f;
#pragma unroll
    for (int k = 0; k < D_CONV; ++k) {
        const int li = l - (D_CONV - 1) + k;
        if (li >= 0) s += w[k] * xrow[li];
    }
    const float sig = 1.0f / (1.0f + __expf(-s));
    xc[((long long)b * D_INNER + d) * L_SEQ + l] = s * sig;
}

// ---------------------------------------------------------------------------
// Selective scan, fused with  y2 = (y + u*D) * silu(z).
// One lane per (b, d) channel; h[16] + A-row in registers; B_t/C_t chunks
// staged in LDS (async b128 when available) and broadcast-read each step.
// Writes y2 over the dead x-half of the xz buffer.
// ---------------------------------------------------------------------------
#define SCAN_CHUNK 64
#define SCAN_STR   (SCAN_CHUNK + 4)   // 68 floats: 272B rows, 16B-aligned, bank-spread

__global__ void scan_kernel(const float* __restrict__ delta,
                            const float* __restrict__ xc,
                            const float* __restrict__ x_dbl,
                            float* __restrict__ xz,      // z read, y2 written (x half)
                            const float* __restrict__ A_log,
                            const float* __restrict__ Dvec)
{
    const int b = blockIdx.y;
    const int d = blockIdx.x * blockDim.x + threadIdx.x;

    __shared__ __align__(16) float Bsh[D_STATE * SCAN_STR];
    __shared__ __align__(16) float Csh[D_STATE * SCAN_STR];

    float Arow[D_STATE];
#pragma unroll
    for (int n = 0; n < D_STATE; ++n)
        Arow[n] = -__expf(A_log[(long long)d * D_STATE + n]);

    float h[D_STATE];
#pragma unroll
    for (int n = 0; n < D_STATE; ++n) h[n] = 0.f;

    const float Dd = Dvec[d];

    const float* __restrict__ Bg =
        x_dbl + (long long)b * (DT_RANK + 2 * D_STATE) * L_SEQ + (long long)DT_RANK * L_SEQ;
    const float* __restrict__ Cg = Bg + (long long)D_STATE * L_SEQ;

    const long long chanBase = ((long long)b * D_INNER + d) * L_SEQ;
    const long long zBase    = ((long long)b * (2 * D_INNER) + D_INNER + d) * L_SEQ;
    const long long yBase    = ((long long)b * (2 * D_INNER) + d) * L_SEQ;

    // 256 threads load 16 rows x 16 float4 per buffer: one b128 per thread.
    const int ln = threadIdx.x >> 4;          // state row n
    const int lc = (threadIdx.x & 15) * 4;    // l-offset (float4)

    for (int l0 = 0; l0 < L_SEQ; l0 += SCAN_CHUNK) {
        if (HAVE_ASYNC_LDS) {
            async_copy16(&Bg[(long long)ln * L_SEQ + l0 + lc], &Bsh[ln * SCAN_STR + lc]);
            async_copy16(&Cg[(long long)ln * L_SEQ + l0 + lc], &Csh[ln * SCAN_STR + lc]);
            wait_async_all();
        } else {
            *(float4*)&Bsh[ln * SCAN_STR + lc] =
                *(const float4*)&Bg[(long long)ln * L_SEQ + l0 + lc];
            *(float4*)&Csh[ln * SCAN_STR + lc] =
                *(const float4*)&Cg[(long long)ln * L_SEQ + l0 + lc];
        }
        __syncthreads();

        for (int c = 0; c < SCAN_CHUNK; ++c) {
            const float dt = delta[chanBase + l0 + c];
            const float u  = xc[chanBase + l0 + c];
            const float du = dt * u;
            float y = 0.f;
#pragma unroll
            for (int n = 0; n < D_STATE; ++n) {
                const float dA = __expf(dt * Arow[n]);
                h[n] = dA * h[n] + du * Bsh[n * SCAN_STR + c];
                y += h[n] * Csh[n * SCAN_STR + c];
            }
            const float zv  = xz[zBase + l0 + c];
            const float sig = 1.0f / (1.0f + __expf(-zv));
            xz[yBase + l0 + c] = (y + u * Dd) * (zv * sig);
        }
        __syncthreads();
    }
}

// ---------------------------------------------------------------------------
// Launcher
// ---------------------------------------------------------------------------
extern "C" void kernel_launch(void* const* d_in, const int* in_sizes, int n_in,
                              void* d_out, int out_size, void* d_ws, size_t ws_size,
                              hipStream_t stream)
{
    const float* hidden    = (const float*)d_in[0]; // (L, B, D_MODEL)
    const float* in_proj_w = (const float*)d_in[1]; // (2*D_INNER, D_MODEL)
    const float* conv_w    = (const float*)d_in[2]; // (D_INNER, D_CONV)
    const float* x_proj_w  = (const float*)d_in[3]; // (DT_RANK+2*D_STATE, D_INNER)
    const float* dt_proj_w = (const float*)d_in[4]; // (D_INNER, DT_RANK)
    const float* dt_proj_b = (const float*)d_in[5]; // (D_INNER,)
    const float* A_log     = (const float*)d_in[6]; // (D_INNER, D_STATE)
    const float* Dvec      = (const float*)d_in[7]; // (D_INNER,)
    const float* out_proj_w= (const float*)d_in[8]; // (D_MODEL, D_INNER)
    float*       out       = (float*)d_out;         // (L, B, D_MODEL)

    float* ws = (float*)d_ws;
    const long long XZ_SZ  = (long long)BATCH * 2 * D_INNER * L_SEQ;
    const long long XC_SZ  = (long long)BATCH * D_INNER * L_SEQ;
    const long long DBL_SZ = (long long)BATCH * (DT_RANK + 2 * D_STATE) * L_SEQ;
    float* xz    = ws;
    float* xc    = xz  + XZ_SZ;
    float* x_dbl = xc  + XC_SZ;
    float* delta = x_dbl + DBL_SZ;

    const dim3 blk(128, 1, 1); // 4 waves (wave32)

    // 1) in_proj: xz[b][e][l] = sum_d W[e,d] * H[l,b,d]   (B contiguous along k)
    {
        const int M = 2 * D_INNER, K = D_MODEL;
        dim3 grd(M / 64, L_SEQ / 64, BATCH);
        gemm_wmma_f32<false, true><<<grd, blk, 0, stream>>>(
            in_proj_w, hidden, xz, nullptr,
            M, K, /*lda=*/D_MODEL,
            /*bbs=*/D_MODEL, /*ldbk=*/1, /*ldbn=*/(long long)BATCH * D_MODEL,
            /*obs=*/(long long)2 * D_INNER * L_SEQ, /*oms=*/L_SEQ, /*ons=*/1);
    }

    // 2) depthwise conv + SiLU  -> xc
    {
        const long long total = (long long)BATCH * D_INNER * L_SEQ;
        conv_silu_kernel<<<(unsigned)((total + 255) / 256), 256, 0, stream>>>(
            xz, conv_w, xc);
    }

    // 3) x_proj: x_dbl[b][e][l] = sum_d Wp[e,d] * xc[b][d][l]   (M=160, ragged)
    {
        const int M = DT_RANK + 2 * D_STATE, K = D_INNER;
        dim3 grd((M + 63) / 64, L_SEQ / 64, BATCH);
        gemm_wmma_f32<false, false><<<grd, blk, 0, stream>>>(
            x_proj_w, xc, x_dbl, nullptr,
            M, K, /*lda=*/D_INNER,
            /*bbs=*/(long long)D_INNER * L_SEQ, /*ldbk=*/L_SEQ, /*ldbn=*/1,
            /*obs=*/(long long)M * L_SEQ, /*oms=*/L_SEQ, /*ons=*/1);
    }

    // 4) dt_proj + bias + softplus -> delta[b][d][l]
    {
        const int M = D_INNER, K = DT_RANK;
        dim3 grd(M / 64, L_SEQ / 64, BATCH);
        gemm_wmma_f32<true, false><<<grd, blk, 0, stream>>>(
            dt_proj_w, x_dbl, delta, dt_proj_b,
            M, K, /*lda=*/DT_RANK,
            /*bbs=*/(long long)(DT_RANK + 2 * D_STATE) * L_SEQ, /*ldbk=*/L_SEQ, /*ldbn=*/1,
            /*obs=*/(long long)D_INNER * L_SEQ, /*oms=*/L_SEQ, /*ons=*/1);
    }

    // 5) selective scan + fused (y + u*D) * silu(z), y2 written into xz x-half
    {
        dim3 grd(D_INNER / 256, BATCH, 1);
        scan_kernel<<<grd, 256, 0, stream>>>(delta, xc, x_dbl, xz, A_log, Dvec);
    }

    // 6) out_proj: out[l][b][m] = sum_d Wo[m,d] * y2[b][d][l]
    {
        const int M = D_MODEL, K = D_INNER;
        dim3 grd(M / 64, L_SEQ / 64, BATCH);
        gemm_wmma_f32<false, false><<<grd, blk, 0, stream>>>(
            out_proj_w, xz, out, nullptr,
            M, K, /*lda=*/D_INNER,
            /*bbs=*/(long long)2 * D_INNER * L_SEQ, /*ldbk=*/L_SEQ, /*ldbn=*/1,
            /*obs=*/D_MODEL, /*oms=*/1, /*ons=*/(long long)BATCH * D_MODEL);
    }
}